// RoiPooling_33440615366842
// MI455X (gfx1250) — compile-verified
//
#include <hip/hip_runtime.h>
#include <cmath>
#include <cfloat>

// ROI max pooling, CDNA5 (gfx1250).
// Grid: (1600 regions, 16 batches). Block: 128 threads (4 wave32s).
// Each thread owns 4 contiguous channels (float4) of the 512-channel vector.
// Memory-bound op: stream the spatial window, fmax-accumulate.
// CDNA5 path: async global->LDS copy engine (GLOBAL_LOAD_ASYNC_TO_LDS_B128,
// ASYNCcnt) with a depth-4 software pipeline; each lane consumes only the
// bytes it staged, so per-wave s_wait_asynccnt suffices (no barriers).
// ASYNCcnt completes in order, so s_wait_asynccnt(DEPTH-1) after issuing
// point i+DEPTH-1 guarantees point i's 16B are resident in LDS.

#ifndef __has_builtin
#define __has_builtin(x) 0
#endif

#if __has_builtin(__builtin_amdgcn_global_load_async_to_lds_b128) && \
    __has_builtin(__builtin_amdgcn_s_wait_asynccnt)
#define ROI_ASYNC_LDS 1
#else
#define ROI_ASYNC_LDS 0
#endif

#define ROI_NUM_ROIS     32
#define ROI_REG_PER_ROI  50
#define ROI_NUM_REGIONS  (ROI_NUM_ROIS * ROI_REG_PER_ROI)   // 1600
#define ROI_BATCH        16
#define ROI_H            96
#define ROI_W            96
#define ROI_C            512
#define ROI_THREADS      128   // 128 * float4 = 512 channels
#define ROI_DEPTH        4     // async pipeline depth (power of 2)

#if ROI_ASYNC_LDS
// Builtin signature (probe-confirmed): arg0 = pointer to 16-byte int vector
// in device (AS1) address space; arg1 = LDS (AS3) side; imm offset; imm cpol.
typedef int roi_v4i __attribute__((vector_size(4 * sizeof(int))));
typedef __attribute__((address_space(1))) roi_v4i* roi_gp_v4i;
typedef __attribute__((address_space(3))) roi_v4i* roi_lp_v4i;

__device__ __forceinline__ void roi_async_cp16(const float* g, float* l) {
    // 16 bytes per lane: global -> LDS, tracked by ASYNCcnt.
    __builtin_amdgcn_global_load_async_to_lds_b128(
        (roi_gp_v4i)(g), (roi_lp_v4i)(l), /*offset=*/0, /*cpol=*/0);
}
#endif

__global__ __launch_bounds__(ROI_THREADS)
void RoiPooling_33440615366842_kernel(const float* __restrict__ fm,
                                      const int*   __restrict__ rois,
                                      float*       __restrict__ out)
{
    const int r = blockIdx.x;   // region index 0..1599
    const int b = blockIdx.y;   // batch 0..15
    const int t = threadIdx.x;  // 0..127

    const int roi = r / ROI_REG_PER_ROI;
    const int q   = r - roi * ROI_REG_PER_ROI;

    // Decode (pool level p, ix, jy) from q. Region order: levels [1,2,3,6],
    // inner loops: for ix in range(p): for jy in range(p).
    int p, ix, jy;
    if (q == 0)      { p = 1; ix = 0;          jy = 0; }
    else if (q < 5)  { p = 2; int u = q - 1;  ix = u >> 1;  jy = u & 1; }
    else if (q < 14) { p = 3; int u = q - 5;  ix = u / 3;   jy = u - (u / 3) * 3; }
    else             { p = 6; int u = q - 14; ix = u / 6;   jy = u - (u / 6) * 6; }

    const int x = rois[roi * 4 + 0];
    const int y = rois[roi * 4 + 1];
    const int w = rois[roi * 4 + 2];
    const int h = rois[roi * 4 + 3];

    // Reference quirk: x-extent uses col_length = h/p, y-extent uses
    // row_length = w/p. Python round() == rint (round-half-even) on doubles.
    const double cl = (double)h / (double)p;
    const double rl = (double)w / (double)p;
    const double ax = (double)x + (double)ix * cl;
    const double ay = (double)y + (double)jy * rl;
    const int x1 = (int)rint(ax);
    const int x2 = (int)rint(ax + cl);
    const int y1 = (int)rint(ay);
    const int y2 = (int)rint(ay + rl);

    const int Wr = x2 - x1;          // always >= 1 (cl, rl > 1)
    const int N  = (y2 - y1) * Wr;   // spatial points in this window

    const size_t coff = (size_t)t * 4;
    const float* fmb = fm + (size_t)b * (ROI_H * ROI_W * ROI_C) + coff;

    float4 acc;
    acc.x = acc.y = acc.z = acc.w = -FLT_MAX;

#if ROI_ASYNC_LDS
    // Depth-4 async pipeline: keep up to 3 spatial points in flight while
    // reducing the oldest. Slot for point i is (i & 3).
    __shared__ float buf[ROI_DEPTH][ROI_THREADS * 4];   // 4 x 2KB = 8KB
    float* const myslot0 = &buf[0][t * 4];              // slot k = myslot0 + k*512

    int py = y1, px = x1;  // prefetch cursor
    const int pre = (N < (ROI_DEPTH - 1)) ? N : (ROI_DEPTH - 1);
    for (int k = 0; k < pre; ++k) {
        roi_async_cp16(fmb + ((size_t)py * ROI_W + px) * ROI_C,
                       myslot0 + (size_t)k * (ROI_THREADS * 4));
        ++px; if (px == x2) { px = x1; ++py; }
    }

    for (int i = 0; i < N; ++i) {
        const int j = i + (ROI_DEPTH - 1);
        if (j < N) {
            roi_async_cp16(fmb + ((size_t)py * ROI_W + px) * ROI_C,
                           myslot0 + (size_t)(j & (ROI_DEPTH - 1)) * (ROI_THREADS * 4));
            ++px; if (px == x2) { px = x1; ++py; }
            // In-order completion: <= DEPTH-1 outstanding (the newest 3)
            // implies point i's copy is done.
            __builtin_amdgcn_s_wait_asynccnt(ROI_DEPTH - 1);
        } else {
            __builtin_amdgcn_s_wait_asynccnt(0);   // drain tail
        }
        __asm__ __volatile__("" ::: "memory");     // keep ds_load after wait
        const float* s = myslot0 + (size_t)(i & (ROI_DEPTH - 1)) * (ROI_THREADS * 4);
        float4 v = *(const float4*)s;
        acc.x = fmaxf(acc.x, v.x);
        acc.y = fmaxf(acc.y, v.y);
        acc.z = fmaxf(acc.z, v.z);
        acc.w = fmaxf(acc.w, v.w);
    }
#else
    // Fallback: direct coalesced float4 streaming loads.
    for (int yy = y1; yy < y2; ++yy) {
        const float* rowp = fmb + (size_t)yy * (ROI_W * ROI_C);
        for (int xx = x1; xx < x2; ++xx) {
            float4 v = *(const float4*)(rowp + (size_t)xx * ROI_C);
            acc.x = fmaxf(acc.x, v.x);
            acc.y = fmaxf(acc.y, v.y);
            acc.z = fmaxf(acc.z, v.z);
            acc.w = fmaxf(acc.w, v.w);
        }
    }
#endif

    // Output layout: (batch, region, channel) = (16, 1600, 512).
    *(float4*)(out + ((size_t)b * ROI_NUM_REGIONS + r) * ROI_C + coff) = acc;
}

extern "C" void kernel_launch(void* const* d_in, const int* in_sizes, int n_in,
                              void* d_out, int out_size, void* d_ws, size_t ws_size,
                              hipStream_t stream) {
    (void)in_sizes; (void)n_in; (void)d_ws; (void)ws_size; (void)out_size;
    const float* fm   = (const float*)d_in[0];
    const int*   rois = (const int*)d_in[1];
    float*       out  = (float*)d_out;

    dim3 grid(ROI_NUM_REGIONS, ROI_BATCH, 1);
    RoiPooling_33440615366842_kernel<<<grid, ROI_THREADS, 0, stream>>>(fm, rois, out);
}